// Convolution6D_46325517254753
// MI455X (gfx1250) — compile-verified
//
#include <hip/hip_runtime.h>
#include <math.h>

typedef float v2f __attribute__((ext_vector_type(2)));
typedef float v8f __attribute__((ext_vector_type(8)));

#define N_PTS 8192
#define T_ER  15
#define P_RAD 5

static constexpr double D_PI = 3.14159265358979323846;

// ---------------- Clebsch-Gordan (device, double precision) ----------------
__device__ double dfact(int n) {
  double r = 1.0;
  for (int i = 2; i <= n; ++i) r *= (double)i;
  return r;
}

__device__ double cgc(int j1, int m1, int j2, int m2, int j3, int m3) {
  if (m1 + m2 != m3) return 0.0;
  if (j3 < abs(j1 - j2) || j3 > j1 + j2) return 0.0;
  if (abs(m1) > j1 || abs(m2) > j2 || abs(m3) > j3) return 0.0;
  double pre = sqrt((2.0 * j3 + 1.0) * dfact(j3 + j1 - j2) * dfact(j3 - j1 + j2) *
                    dfact(j1 + j2 - j3) / dfact(j1 + j2 + j3 + 1));
  pre *= sqrt(dfact(j3 + m3) * dfact(j3 - m3) * dfact(j1 - m1) * dfact(j1 + m1) *
              dfact(j2 - m2) * dfact(j2 + m2));
  int kmin = max(0, max(j2 - j3 - m1, j1 - j3 + m2));
  int kmax = min(j1 + j2 - j3, min(j1 - m1, j2 + m2));
  double s = 0.0;
  for (int k = kmin; k <= kmax; ++k) {
    double d = dfact(k) * dfact(j1 + j2 - j3 - k) * dfact(j1 - m1 - k) *
               dfact(j2 + m2 - k) * dfact(j3 - j2 + m1 + k) * dfact(j3 - j1 - m2 + k);
    s += ((k & 1) ? -1.0 : 1.0) / d;
  }
  return pre * s;
}

// H[quad][l][l2][lidx][k][l1m] : combined +/- c * (T1 +/- T2 +/- T3) tables.
// quad 0: o_r <- x_r (uses w_r)   c*(T1+T2+T3)
// quad 1: o_r <- x_i (uses w_i)  -c*(T1-T2+T3)   (w_i m=0 masked)
// quad 2: o_i <- x_r (uses w_i)   c*(T1-T2-T3)   (w_i m=0 masked)
// quad 3: o_i <- x_i (uses w_r)   c*(T1+T2-T3)
__global__ void build_H(float* __restrict__ H) {
  int idx = blockIdx.x * blockDim.x + threadIdx.x;
  if (idx >= 1024) return;
  int k    = idx & 3;
  int lidx = (idx >> 2) & 3;
  int l2   = (idx >> 4) & 3;
  int l    = (idx >> 6) & 3;
  int quad = (idx >> 8) & 3;
  for (int l1m = 0; l1m < 10; ++l1m) {
    int l1 = (l1m >= 1) + (l1m >= 3) + (l1m >= 6);
    int m  = l1m - (l1 * (l1 + 1)) / 2;
    double val = 0.0;
    bool valid = (lidx <= l) && (k <= l2) && (m <= l1) &&
                 (l2 >= abs(l - l1)) && (l2 <= min(3, l + l1));
    if (valid) {
      double c = 8.0 * D_PI * D_PI / (2.0 * l1 + 1.0)
               * sqrt((2.0 * l + 1.0) * (2.0 * l1 + 1.0) / (4.0 * D_PI * (2.0 * l2 + 1.0)))
               * cgc(l, 0, l1, 0, l2, 0);
      double t1 = 0.0, t2 = 0.0, t3 = 0.0;
      if (lidx + m == k)
        t1 = cgc(l, lidx, l1, m, l2, k) * ((m & 1) ? -1.0 : 1.0);
      if (m > 0 && (lidx - m) == k)
        t2 = cgc(l, lidx, l1, -m, l2, k) * ((l1 & 1) ? -1.0 : 1.0);
      if (m > 0 && (m - lidx) == k && k > 0)
        t3 = ((k & 1) ? -1.0 : 1.0) * (((l1 + l2) & 1) ? -1.0 : 1.0) *
             cgc(l, lidx, l1, -m, l2, lidx - m);
      double v;
      if      (quad == 0) v =  c * (t1 + t2 + t3);
      else if (quad == 1) v = -c * (t1 - t2 + t3);
      else if (quad == 2) v =  c * (t1 - t2 - t3);
      else                v =  c * (t1 + t2 - t3);
      if ((quad == 1 || quad == 2) && m == 0) v = 0.0;  // w_i m=0 masked
      val = v;
    }
    H[((((quad * 4 + l) * 4 + l2) * 4 + lidx) * 4 + k) * 10 + l1m] = (float)val;
  }
}

// W_eff[p][r=4800][col=320], stored row-pair interleaved:
//   elem(r,c) -> p*1536000 + (r>>1)*640 + c*2 + (r&1)
// r = part*2400 + K10*240 + t*16 + i ; col = opart*160 + colb*16 + j
__global__ __launch_bounds__(256) void build_Weff(
    float* __restrict__ W, const float* __restrict__ H,
    const float* __restrict__ w0r, const float* __restrict__ w0i,
    const float* __restrict__ w1r, const float* __restrict__ w1i,
    const float* __restrict__ w2r, const float* __restrict__ w2i,
    const float* __restrict__ w3r, const float* __restrict__ w3i)
{
  int idx = blockIdx.x * blockDim.x + threadIdx.x;
  if (idx >= 7680000) return;
  const float* wr[4] = {w0r, w1r, w2r, w3r};
  const float* wi[4] = {w0i, w1i, w2i, w3i};

  int p    = idx / 1536000;
  int rem  = idx - p * 1536000;
  int r    = rem / 320;
  int col  = rem - r * 320;

  int part = r / 2400;
  int rr   = r - part * 2400;
  int K10  = rr / 240;
  int tt   = rr - K10 * 240;
  int t    = tt >> 4;
  int i    = tt & 15;
  int l2   = (K10 >= 1) + (K10 >= 3) + (K10 >= 6);
  int k    = K10 - (l2 * (l2 + 1)) / 2;

  int opart = col / 160;
  int cc    = col - opart * 160;
  int colb  = cc >> 4;
  int j     = cc & 15;
  int l     = (colb >= 1) + (colb >= 3) + (colb >= 6);
  int lidx  = colb - (l * (l + 1)) / 2;

  int quad = opart * 2 + part;
  const int hbase = ((((quad * 4 + l) * 4 + l2) * 4 + lidx) * 4 + k) * 10;

  float val = 0.0f;
#pragma unroll
  for (int l1m = 0; l1m < 10; ++l1m) {
    const int l1 = (l1m >= 1) + (l1m >= 3) + (l1m >= 6);
    const int m  = l1m - (l1 * (l1 + 1)) / 2;
    float h = H[hbase + l1m];
    const float* w = (quad == 0 || quad == 3) ? wr[l1] : wi[l1];
    val += h * w[m * 19200 + t * 1280 + p * 256 + i * 16 + j];
  }
  W[(size_t)p * 1536000 + (size_t)(r >> 1) * 640 + col * 2 + (r & 1)] = val;
}

// Main batched GEMM: per p, [8192 x 4800] x [4800 x 320] via f32 WMMA 16x16x4.
// 8 waves/WG: wave = (row-group rg 0..3, column-half ch 0..1).
// Wave tile: 16 rows x 160 cols = 10 accumulators of v8f.
__global__ __launch_bounds__(256) void conv6d_wmma(
    const float* __restrict__ x0r, const float* __restrict__ x0i,
    const float* __restrict__ x1r, const float* __restrict__ x1i,
    const float* __restrict__ x2r, const float* __restrict__ x2i,
    const float* __restrict__ x3r, const float* __restrict__ x3i,
    const float* __restrict__ Weff,
    const float* __restrict__ b0r, const float* __restrict__ b0i,
    const float* __restrict__ b1r, const float* __restrict__ b1i,
    const float* __restrict__ b2r, const float* __restrict__ b2i,
    const float* __restrict__ b3r, const float* __restrict__ b3i,
    float* __restrict__ out)
{
  const int tid   = threadIdx.x;
  const int wave  = tid >> 5;
  const int lane  = tid & 31;
  const int rg    = wave >> 1;
  const int ch    = wave & 1;
  const int laneM = lane & 15;
  const int khalf = lane >> 4;
  const int p     = blockIdx.y;
  const int n0    = blockIdx.x * 64 + rg * 16;
  const int nrow  = n0 + laneM;

  const float* xs[8] = {x0r, x1r, x2r, x3r, x0i, x1i, x2i, x3i};

  v8f acc[10];
  const v8f vzero = {0.f, 0.f, 0.f, 0.f, 0.f, 0.f, 0.f, 0.f};
#pragma unroll
  for (int ct = 0; ct < 10; ++ct) acc[ct] = vzero;

  // lane-resolved base into interleaved W_eff
  const float* Wlane = Weff + (size_t)p * 1536000 + ch * 320 + laneM * 2;

#pragma unroll
  for (int part = 0; part < 2; ++part) {
#pragma unroll
    for (int K10 = 0; K10 < 10; ++K10) {
      const int l2 = (K10 >= 1) + (K10 >= 3) + (K10 >= 6);
      const int k  = K10 - (l2 * (l2 + 1)) / 2;
      const int strideT = (l2 + 1) * 80;
      const float* xb = xs[part * 4 + l2]
                      + (size_t)nrow * (size_t)(T_ER * strideT)
                      + (size_t)(k * 80 + p * 16 + khalf * 2);
      const int segBase = part * 150 + K10 * 15;
      for (int t = 0; t < 15; ++t) {
        const float* xa = xb + t * strideT;
        const float* wb = Wlane + (size_t)(segBase + t) * 5120 + khalf * 640;
#pragma unroll
        for (int ksub = 0; ksub < 4; ++ksub) {
          // A fragment (16x4 f32): v0 = K {0|2}, v1 = K {1|3} per half-wave
          v2f a = *(const v2f*)(xa + ksub * 4);
          const float* wk = wb + ksub * 1280;
#pragma unroll
          for (int ct = 0; ct < 10; ++ct) {
            v2f b = *(const v2f*)(wk + ct * 32);
            acc[ct] = __builtin_amdgcn_wmma_f32_16x16x4_f32(
                false, a, false, b, (short)0, acc[ct], false, false);
          }
        }
      }
    }
  }

  // -------- epilogue: bias + store into concatenated output ----------------
  const float* bs_r[4] = {b0r, b1r, b2r, b3r};
  const float* bs_i[4] = {b0i, b1i, b2i, b3i};
  const size_t offR[4] = {0u, 1310720u, 3932160u, 7864320u};
  const size_t offI[4] = {655360u, 2621440u, 5898240u, 10485760u};

#pragma unroll
  for (int ct = 0; ct < 10; ++ct) {
    const int l    = (ct >= 1) + (ct >= 3) + (ct >= 6);
    const int lidx = ct - (l * (l + 1)) / 2;
    const float* bp = ch ? bs_i[l] : bs_r[l];
    float bias = bp[lidx * 80 + p * 16 + laneM];
    if (ch && lidx == 0) bias = 0.0f;  // b_i lidx==0 masked
    float* op = out + (ch ? offI[l] : offR[l]) + (size_t)(lidx * 80 + p * 16 + laneM);
    const int strideN = (l + 1) * 80;
#pragma unroll
    for (int v = 0; v < 8; ++v) {
      const int n = n0 + v + khalf * 8;  // C/D layout: VGPR v = rows v / v+8
      op[(size_t)n * (size_t)strideN] = acc[ct][v] + bias;
    }
  }
}

extern "C" void kernel_launch(void* const* d_in, const int* in_sizes, int n_in,
                              void* d_out, int out_size, void* d_ws, size_t ws_size,
                              hipStream_t stream) {
  (void)in_sizes; (void)n_in; (void)out_size; (void)ws_size;
  float* H    = (float*)d_ws;            // 10,240 floats
  float* Weff = (float*)d_ws + 16384;    // 7,680,000 floats (~30.7 MB)

  build_H<<<4, 256, 0, stream>>>(H);
  build_Weff<<<30000, 256, 0, stream>>>(
      Weff, H,
      (const float*)d_in[8],  (const float*)d_in[9],
      (const float*)d_in[10], (const float*)d_in[11],
      (const float*)d_in[12], (const float*)d_in[13],
      (const float*)d_in[14], (const float*)d_in[15]);
  conv6d_wmma<<<dim3(128, 5), 256, 0, stream>>>(
      (const float*)d_in[0], (const float*)d_in[1],
      (const float*)d_in[2], (const float*)d_in[3],
      (const float*)d_in[4], (const float*)d_in[5],
      (const float*)d_in[6], (const float*)d_in[7],
      Weff,
      (const float*)d_in[16], (const float*)d_in[17],
      (const float*)d_in[18], (const float*)d_in[19],
      (const float*)d_in[20], (const float*)d_in[21],
      (const float*)d_in[22], (const float*)d_in[23],
      (float*)d_out);
}